// EarthMoverDistance_37649683317563
// MI455X (gfx1250) — compile-verified
//
#include <hip/hip_runtime.h>

// Sinkhorn EMD on gfx1250 (MI455X).
//  - Cost matrix via V_WMMA_F32_16X16X4_F32 (K=3 padded to 4, exact f32).
//  - C (32 MB) lives in d_ws and stays L2-resident (192 MB L2) across the
//    40 logsumexp sweeps + final contraction.

typedef __attribute__((ext_vector_type(2))) float v2f;
typedef __attribute__((ext_vector_type(8))) float v8f;

#define B_      8
#define N_      1024
#define M_      1024
#define EPS_    0.05f
#define INV_EPS_ 20.0f
#define LOG_N   6.931471805599453f   /* ln(1024) */

// ---------------------------------------------------------------------------
// Kernel 1: C[b,i,j] = max(||x_i||^2 + ||y_j||^2 - 2 x_i.y_j, 0)
// One wave per 16x16 tile. xy dot-products via v_wmma_f32_16x16x4_f32.
// A-matrix f32 16x4 layout (ISA 7.12.2): lanes 0-15 -> K=0(V0),K=1(V1);
// lanes 16-31 -> K=2(V0),K=3(V1). B-matrix 4x16 mirrored. D: lane<16 holds
// rows M=v (VGPR v), N=lane; lane>=16 rows M=v+8, N=lane-16.
// ---------------------------------------------------------------------------
__global__ void emd_cost_wmma(const float* __restrict__ xyz1,
                              const float* __restrict__ xyz2,
                              float* __restrict__ C) {
  const int wave = (blockIdx.x * blockDim.x + threadIdx.x) >> 5;
  const int lane = threadIdx.x & 31;

  const int tiles_per_b = (N_ / 16) * (M_ / 16);       // 4096
  const int b  = wave / tiles_per_b;
  const int t  = wave % tiles_per_b;
  const int r0 = (t >> 6) << 4;                        // tile row base
  const int c0 = (t & 63) << 4;                        // tile col base

  const float* X = xyz1 + (size_t)b * N_ * 3;
  const float* Y = xyz2 + (size_t)b * M_ * 3;

  const int half = lane >> 4;                          // K-half select
  const int mn   = lane & 15;

  const float* xr = X + (size_t)(r0 + mn) * 3;
  const float* yr = Y + (size_t)(c0 + mn) * 3;
  const float x0 = xr[0], x1 = xr[1], x2c = xr[2];
  const float y0 = yr[0], y1 = yr[1], y2c = yr[2];

  v2f a, bmat;                                         // branchless (EXEC all-1)
  a.x    = half ? x2c : x0;
  a.y    = half ? 0.0f : x1;
  bmat.x = half ? y2c : y0;
  bmat.y = half ? 0.0f : y1;

  v8f acc = {0.f, 0.f, 0.f, 0.f, 0.f, 0.f, 0.f, 0.f};
  acc = __builtin_amdgcn_wmma_f32_16x16x4_f32(
      /*neg_a=*/false, a, /*neg_b=*/false, bmat,
      /*c_mod=*/(short)0, acc, /*reuse_a=*/false, /*reuse_b=*/false);

  const int   mbase = half << 3;
  const float y2    = y0 * y0 + y1 * y1 + y2c * y2c;   // our column's ||y||^2

  float* Cout = C + (size_t)b * N_ * M_ + (size_t)(r0 + mbase) * M_ + (c0 + mn);
#pragma unroll
  for (int v = 0; v < 8; ++v) {
    const float* xv = X + (size_t)(r0 + mbase + v) * 3;
    const float  x2 = xv[0] * xv[0] + xv[1] * xv[1] + xv[2] * xv[2];
    float cij = x2 + y2 - 2.0f * acc[v];
    cij = cij > 0.0f ? cij : 0.0f;
    Cout[(size_t)v * M_] = cij;
  }
}

// ---------------------------------------------------------------------------
// Kernel 2: f_i = -eps * lse_j( (g_j - C_ij)/eps - lnN ).  Wave per row,
// lanes stride the columns (coalesced), online logsumexp + shfl_xor merge.
// ---------------------------------------------------------------------------
__global__ void emd_f_kernel(const float* __restrict__ C,
                             const float* __restrict__ g,
                             float* __restrict__ f) {
  const int wave = (blockIdx.x * blockDim.x + threadIdx.x) >> 5;
  const int lane = threadIdx.x & 31;
  const int b = wave / N_;
  const int i = wave % N_;

  const float* Crow = C + (size_t)b * N_ * M_ + (size_t)i * M_;
  const float* gb   = g + b * M_;

  float mx = -3.0e38f, sum = 0.0f;
  for (int k = 0; k < M_ / 32; ++k) {
    const int   j = lane + (k << 5);
    const float v = (gb[j] - Crow[j]) * INV_EPS_ - LOG_N;
    if (v > mx) { sum = sum * __expf(mx - v) + 1.0f; mx = v; }
    else        { sum += __expf(v - mx); }
  }
#pragma unroll
  for (int off = 16; off > 0; off >>= 1) {
    const float mo = __shfl_xor(mx,  off, 32);
    const float so = __shfl_xor(sum, off, 32);
    const float m  = mx > mo ? mx : mo;
    sum = sum * __expf(mx - m) + so * __expf(mo - m);
    mx  = m;
  }
  if (lane == 0) f[b * N_ + i] = -EPS_ * (mx + __logf(sum));
}

// ---------------------------------------------------------------------------
// Kernel 3: g_j = -eps * lse_i( (f_i - C_ij)/eps - lnN ).  Thread per column
// (consecutive threads -> coalesced row reads); f_i is wave-uniform.
// ---------------------------------------------------------------------------
__global__ void emd_g_kernel(const float* __restrict__ C,
                             const float* __restrict__ f,
                             float* __restrict__ g) {
  const int t = blockIdx.x * blockDim.x + threadIdx.x;   // B*M threads
  const int b = t / M_;
  const int j = t % M_;

  const float* Cb = C + (size_t)b * N_ * M_;
  const float* fb = f + b * N_;

  float mx = -3.0e38f, sum = 0.0f;
  for (int i = 0; i < N_; ++i) {
    const float v = (fb[i] - Cb[(size_t)i * M_ + j]) * INV_EPS_ - LOG_N;
    if (v > mx) { sum = sum * __expf(mx - v) + 1.0f; mx = v; }
    else        { sum += __expf(v - mx); }
  }
  g[t] = -EPS_ * (mx + __logf(sum));
}

// ---------------------------------------------------------------------------
// Kernel 4: emd = sum_ij exp((f_i+g_j-C_ij)/eps - 2 lnN) * C_ij, mean over B.
// 64 blocks per batch, 16 rows each, coalesced; block reduce + atomicAdd/8.
// ---------------------------------------------------------------------------
__global__ void emd_final_kernel(const float* __restrict__ C,
                                 const float* __restrict__ f,
                                 const float* __restrict__ g,
                                 float* __restrict__ out) {
  __shared__ float red[256];
  const int b  = blockIdx.x >> 6;
  const int rb = blockIdx.x & 63;

  const float* Cb = C + (size_t)b * N_ * M_ + (size_t)rb * 16 * M_;
  const float* fb = f + b * N_ + rb * 16;
  const float* gb = g + b * M_;

  float acc = 0.0f;
  for (int k = 0; k < 64; ++k) {
    const int   idx = (k << 8) + threadIdx.x;            // 0..16383
    const int   row = idx >> 10;
    const int   col = idx & (M_ - 1);
    const float cv  = Cb[idx];
    const float lp  = (fb[row] + gb[col] - cv) * INV_EPS_ - 2.0f * LOG_N;
    acc += __expf(lp) * cv;
  }
  red[threadIdx.x] = acc;
  __syncthreads();
  for (int s = 128; s > 0; s >>= 1) {
    if (threadIdx.x < s) red[threadIdx.x] += red[threadIdx.x + s];
    __syncthreads();
  }
  if (threadIdx.x == 0) atomicAdd(out, red[0] * (1.0f / (float)B_));
}

// ---------------------------------------------------------------------------
extern "C" void kernel_launch(void* const* d_in, const int* in_sizes, int n_in,
                              void* d_out, int out_size, void* d_ws, size_t ws_size,
                              hipStream_t stream) {
  const float* xyz1 = (const float*)d_in[0];
  const float* xyz2 = (const float*)d_in[1];
  float* out = (float*)d_out;

  // ws layout: C (B*N*M floats) | f (B*N) | g (B*M)
  float* C = (float*)d_ws;
  float* f = C + (size_t)B_ * N_ * M_;
  float* g = f + (size_t)B_ * N_;

  hipMemsetAsync(out, 0, sizeof(float), stream);
  hipMemsetAsync(f, 0, (size_t)(B_ * N_ + B_ * M_) * sizeof(float), stream);

  // 1) Cost matrix via WMMA: 8*64*64 tiles, 8 waves/block.
  emd_cost_wmma<<<(B_ * (N_ / 16) * (M_ / 16)) / 8, 256, 0, stream>>>(xyz1, xyz2, C);

  // 2) 20 Sinkhorn iterations (f then g, matching reference scan order).
  for (int it = 0; it < 20; ++it) {
    emd_f_kernel<<<(B_ * N_ * 32) / 256, 256, 0, stream>>>(C, g, f);
    emd_g_kernel<<<(B_ * M_) / 256, 256, 0, stream>>>(C, f, g);
  }

  // 3) Transport-plan contraction -> mean EMD.
  emd_final_kernel<<<B_ * 64, 256, 0, stream>>>(C, f, g, out);
}